// SpacialSeparation_36223754174590
// MI455X (gfx1250) — compile-verified
//
#include <hip/hip_runtime.h>

typedef __attribute__((ext_vector_type(2))) float v2f;
typedef __attribute__((ext_vector_type(8))) float v8f;

#define NROWS     8192
#define DDIM      64
#define TILE      128
#define KC        32          // K chunk held in LDS
#define LSTRIDE   34          // KC + 2 pad (rows land on distinct banks)
#define NWAVES    8

// ---------------- kernel 1: per-row sum of squares ----------------
__global__ void ss_sq_kernel(const float* __restrict__ O, float* __restrict__ sq) {
    int row = blockIdx.x * blockDim.x + threadIdx.x;
    if (row >= NROWS) return;
    const float4* p = (const float4*)(O + (size_t)row * DDIM);
    float s = 0.f;
#pragma unroll
    for (int i = 0; i < DDIM / 4; ++i) {
        float4 v = p[i];
        s += v.x * v.x + v.y * v.y + v.z * v.z + v.w * v.w;
    }
    sq[row] = s;
}

// ---------------- kernel 2: fused WMMA gram tile + epilogue ----------------
__global__ __launch_bounds__(256) void ss_tile_kernel(const float* __restrict__ O,
                                                      const int* __restrict__ labels,
                                                      const float* __restrict__ sq,
                                                      float* __restrict__ partials) {
    __shared__ float As[TILE * LSTRIDE];
    __shared__ float Bs[TILE * LSTRIDE];
    __shared__ float sqI[TILE], sqJ[TILE];
    __shared__ int   labI[TILE], labJ[TILE];
    __shared__ float wsum[NWAVES];

    const int bi  = blockIdx.x, bj = blockIdx.y;
    const int pid = bj * gridDim.x + bi;
    const int tid = threadIdx.x;

    if (bj < bi) {                       // entirely below the j>i diagonal
        if (tid == 0) partials[pid] = 0.f;
        return;
    }
    const int tileI = bi * TILE, tileJ = bj * TILE;

    if (tid < TILE) {
        sqI[tid]  = sq[tileI + tid];
        sqJ[tid]  = sq[tileJ + tid];
        labI[tid] = labels[tileI + tid];
        labJ[tid] = labels[tileJ + tid];
    }

    const int wave = tid >> 5;           // 0..7 (wave32)
    const int lane = tid & 31;
    const int lr   = lane & 15;
    const int hi   = lane >> 4;          // 0 / 1

    const v8f vzero = {};
    v8f acc[8];
#pragma unroll
    for (int n = 0; n < 8; ++n) acc[n] = vzero;

    for (int kc = 0; kc < DDIM / KC; ++kc) {
        __syncthreads();
        // stage A (I rows) and B (J rows) K-chunks into LDS: 128x32 each
#pragma unroll
        for (int i = 0; i < 4; ++i) {
            int idx = tid + i * 256;             // 0..1023
            int row = idx >> 3;                  // 0..127
            int c4  = (idx & 7) << 2;            // 0,4,...,28
            const size_t goff = (size_t)DDIM * row + kc * KC + c4;
            float4 av = *(const float4*)(O + (size_t)tileI * DDIM + goff);
            float4 bv = *(const float4*)(O + (size_t)tileJ * DDIM + goff);
            float* ap = &As[row * LSTRIDE + c4];
            float* bp = &Bs[row * LSTRIDE + c4];
            v2f t;
            t.x = av.x; t.y = av.y; *(v2f*)(ap)     = t;
            t.x = av.z; t.y = av.w; *(v2f*)(ap + 2) = t;
            t.x = bv.x; t.y = bv.y; *(v2f*)(bp)     = t;
            t.x = bv.z; t.y = bv.w; *(v2f*)(bp + 2) = t;
        }
        __syncthreads();

        // WMMA accumulate: wave owns M-strip `wave`, sweeps all 8 N sub-tiles
        for (int k0 = 0; k0 < KC; k0 += 4) {
            const int kk = k0 + 2 * hi;          // per-lane K per 16x4 f32 layout
            v2f a = *(const v2f*)&As[(wave * 16 + lr) * LSTRIDE + kk];
#pragma unroll
            for (int n = 0; n < 8; ++n) {
                v2f b = *(const v2f*)&Bs[(n * 16 + lr) * LSTRIDE + kk];
                acc[n] = __builtin_amdgcn_wmma_f32_16x16x4_f32(
                    false, a, false, b, (short)0, acc[n], false, false);
            }
        }
    }
    __syncthreads();

    // epilogue: d2 = sq_i + sq_j - 2*gram ; dist = d2^(1/4) ; +/- factor
    float s = 0.f;
#pragma unroll
    for (int n = 0; n < 8; ++n) {
#pragma unroll
        for (int v = 0; v < 8; ++v) {
            float g  = acc[n][v];
            int   il = wave * 16 + v + 8 * hi;   // C/D layout: M = v + 8*(lane>=16)
            int   jl = n * 16 + lr;              //             N = lane % 16
            int   gi = tileI + il;
            int   gj = tileJ + jl;
            if (gj > gi) {
                float d2   = fmaxf(sqI[il] + sqJ[jl] - 2.f * g, 0.f);
                float dist = sqrtf(sqrtf(d2));   // (L2^2)^(1/4) == sqrt(L2)
                s += dist * ((labI[il] != labJ[jl]) ? -1.f : 5.f);
            }
        }
    }

    // wave32 shuffle reduction, then cross-wave via LDS
#pragma unroll
    for (int off = 16; off > 0; off >>= 1)
        s += __shfl_xor(s, off, 32);
    if (lane == 0) wsum[wave] = s;
    __syncthreads();
    if (tid == 0) {
        float t = 0.f;
#pragma unroll
        for (int w = 0; w < NWAVES; ++w) t += wsum[w];
        partials[pid] = t;
    }
}

// ---------------- kernel 3: deterministic final reduction ----------------
__global__ void ss_reduce_kernel(const float* __restrict__ partials, int n,
                                 float* __restrict__ out) {
    __shared__ float sm[256];
    int tid = threadIdx.x;
    float s = 0.f;
    for (int i = tid; i < n; i += 256) s += partials[i];
    sm[tid] = s;
    __syncthreads();
    for (int off = 128; off > 0; off >>= 1) {
        if (tid < off) sm[tid] += sm[tid + off];
        __syncthreads();
    }
    if (tid == 0) out[0] = sm[0];
}

extern "C" void kernel_launch(void* const* d_in, const int* in_sizes, int n_in,
                              void* d_out, int out_size, void* d_ws, size_t ws_size,
                              hipStream_t stream) {
    const float* O      = (const float*)d_in[0];
    const int*   labels = (const int*)d_in[1];
    float* sq       = (float*)d_ws;            // 8192 floats
    float* partials = sq + NROWS;              // 4096 floats

    ss_sq_kernel<<<NROWS / 256, 256, 0, stream>>>(O, sq);

    dim3 grid(NROWS / TILE, NROWS / TILE);     // 64 x 64 tile blocks
    ss_tile_kernel<<<grid, 256, 0, stream>>>(O, labels, sq, partials);

    ss_reduce_kernel<<<1, 256, 0, stream>>>(partials,
                                            (NROWS / TILE) * (NROWS / TILE),
                                            (float*)d_out);
}